// SwavModel_55370718380495
// MI455X (gfx1250) — compile-verified
//
#include <hip/hip_runtime.h>
#include <hip/hip_bf16.h>
#include <math.h>

typedef float v2f __attribute__((ext_vector_type(2)));
typedef float v8f __attribute__((ext_vector_type(8)));
typedef int   v4i __attribute__((ext_vector_type(4)));

#define N_ROWS 32768
#define D_DIM  256
#define K_PROT 2048
#define INF_BITS 0x7F800000u

#if __has_builtin(__builtin_amdgcn_global_load_async_to_lds_b128) && \
    __has_builtin(__builtin_amdgcn_s_wait_asynccnt)
#define HAS_ASYNC_LDS 1
#else
#define HAS_ASYNC_LDS 0
#endif

typedef __attribute__((address_space(1))) v4i gv4i;  // global (device) int4
typedef __attribute__((address_space(3))) v4i lv4i;  // LDS int4

// 16-byte global -> LDS copy: async DMA path on gfx1250, staged fallback else.
__device__ __forceinline__ void cp16(const float* g, float* l) {
#if HAS_ASYNC_LDS
    __builtin_amdgcn_global_load_async_to_lds_b128(
        (gv4i*)g, (lv4i*)l, /*offset=*/0, /*cpol=*/0);
#else
    *(float4*)l = *(const float4*)g;
#endif
}

// ---------------------------------------------------------------------------
// Kernel 1: p2[k] = sum_d P[k,d]^2 ; init colmin[k] = +inf
// ---------------------------------------------------------------------------
__global__ __launch_bounds__(256) void prep_kernel(
    const float* __restrict__ P, float* __restrict__ p2,
    unsigned int* __restrict__ colmin)
{
    const int wave = threadIdx.x >> 5;
    const int lane = threadIdx.x & 31;
    const int k = blockIdx.x * 8 + wave;
    const float* pr = P + (size_t)k * D_DIM;

    float s = 0.0f;
    #pragma unroll
    for (int i = 0; i < 2; ++i) {
        float4 v = *(const float4*)(pr + i * 128 + lane * 4);
        s += v.x * v.x + v.y * v.y + v.z * v.z + v.w * v.w;
    }
    s += __shfl_xor(s, 1, 32);
    s += __shfl_xor(s, 2, 32);
    s += __shfl_xor(s, 4, 32);
    s += __shfl_xor(s, 8, 32);
    s += __shfl_xor(s, 16, 32);
    if (lane == 0) {
        p2[k] = s;
        colmin[k] = INF_BITS;
    }
}

// ---------------------------------------------------------------------------
// Kernel 2: per-row z2, invnorm; write out0 (copy) / out1 (normalized);
// init rowmin. One wave per row.
// ---------------------------------------------------------------------------
__global__ __launch_bounds__(256) void rownorm_kernel(
    const float* __restrict__ Z,
    float* __restrict__ out0, float* __restrict__ out1,
    float* __restrict__ z2, float* __restrict__ invn,
    unsigned int* __restrict__ rowmin)
{
    const int wave = threadIdx.x >> 5;
    const int lane = threadIdx.x & 31;
    const int n = blockIdx.x * 8 + wave;
    const float* zr = Z + (size_t)n * D_DIM;

    float4 v0 = *(const float4*)(zr + lane * 4);
    float4 v1 = *(const float4*)(zr + 128 + lane * 4);
    float s = v0.x * v0.x + v0.y * v0.y + v0.z * v0.z + v0.w * v0.w
            + v1.x * v1.x + v1.y * v1.y + v1.z * v1.z + v1.w * v1.w;
    s += __shfl_xor(s, 1, 32);
    s += __shfl_xor(s, 2, 32);
    s += __shfl_xor(s, 4, 32);
    s += __shfl_xor(s, 8, 32);
    s += __shfl_xor(s, 16, 32);

    const float nrm = sqrtf(s);
    const float inv = 1.0f / fmaxf(nrm, 1e-12f);
    if (lane == 0) {
        z2[n] = s;
        invn[n] = inv;
        rowmin[n] = INF_BITS;
    }

    float* o0 = out0 + (size_t)n * D_DIM;
    float* o1 = out1 + (size_t)n * D_DIM;
    *(float4*)(o0 + lane * 4)       = v0;
    *(float4*)(o0 + 128 + lane * 4) = v1;
    float4 w0 = make_float4(v0.x * inv, v0.y * inv, v0.z * inv, v0.w * inv);
    float4 w1 = make_float4(v1.x * inv, v1.y * inv, v1.z * inv, v1.w * inv);
    *(float4*)(o1 + lane * 4)       = w0;
    *(float4*)(o1 + 128 + lane * 4) = w1;
}

// ---------------------------------------------------------------------------
// Kernel 3: fused WMMA f32 GEMM  dot = Z @ P.T  (block tile 128x64)
// Double-buffered async global->LDS pipeline, K-chunk = 32.
// 8 waves; wave w owns rows [w*16, w*16+16) x 64 cols (4 accumulator tiles).
// Epilogue: x_mapped = dot*invnorm[n]; dist = sqrt(max(z2+p2-2dot,0));
//           shuffle-reduced row/col mins -> atomicMin on float bits.
// ---------------------------------------------------------------------------
__global__ __launch_bounds__(256) void gemm_fused_kernel(
    const float* __restrict__ Z, const float* __restrict__ P,
    const float* __restrict__ z2, const float* __restrict__ invn,
    const float* __restrict__ p2,
    float* __restrict__ xmapped,
    unsigned int* __restrict__ rowmin, unsigned int* __restrict__ colmin)
{
    // stride 36 floats = 144B: 16B-aligned rows for B128 async writes,
    // and 36*r mod 64 distinct for r=0..15 -> conflict-free fragment reads.
    __shared__ float zt[2][128][36];
    __shared__ float pt[2][64][36];

    const int tid  = threadIdx.x;
    const int wave = tid >> 5;
    const int lane = tid & 31;
    const int n0 = blockIdx.x * 128;
    const int k0 = blockIdx.y * 64;

    v8f acc[4];
    const v8f zero = {0.f, 0.f, 0.f, 0.f, 0.f, 0.f, 0.f, 0.f};
    #pragma unroll
    for (int t = 0; t < 4; ++t) acc[t] = zero;

    const int m    = lane & 15;          // position within 16
    const int khi  = (lane >> 4) << 1;   // 0 or 2 (k sub-lane split)
    const int arow = wave * 16 + m;

    // issue async copies for one 32-wide K-chunk into buffer `buf`
    auto load_chunk = [&](int d0, int buf) {
        #pragma unroll
        for (int i = 0; i < 4; ++i) {            // Z: 128x32 = 1024 x 16B
            const int linear = tid + i * 256;
            const int r  = linear >> 3;
            const int c4 = (linear & 7) << 2;
            cp16(Z + (size_t)(n0 + r) * D_DIM + d0 + c4, &zt[buf][r][c4]);
        }
        #pragma unroll
        for (int i = 0; i < 2; ++i) {            // P: 64x32 = 512 x 16B
            const int linear = tid + i * 256;
            const int r  = linear >> 3;
            const int c4 = (linear & 7) << 2;
            cp16(P + (size_t)(k0 + r) * D_DIM + d0 + c4, &pt[buf][r][c4]);
        }
    };

    load_chunk(0, 0);  // prime the pipeline

    for (int c = 0; c < 8; ++c) {
        const int buf = c & 1;
        if (c < 7) {
            load_chunk((c + 1) * 32, buf ^ 1);   // prefetch next chunk
#if HAS_ASYNC_LDS
            __builtin_amdgcn_s_wait_asynccnt(6); // this chunk done; next in flight
#endif
        } else {
#if HAS_ASYNC_LDS
            __builtin_amdgcn_s_wait_asynccnt(0);
#endif
        }
        __syncthreads();

        #pragma unroll
        for (int dd = 0; dd < 32; dd += 4) {
            v2f a;
            a.x = zt[buf][arow][dd + khi];
            a.y = zt[buf][arow][dd + khi + 1];
            #pragma unroll
            for (int t = 0; t < 4; ++t) {
                v2f b;
                b.x = pt[buf][t * 16 + m][dd + khi];
                b.y = pt[buf][t * 16 + m][dd + khi + 1];
                acc[t] = __builtin_amdgcn_wmma_f32_16x16x4_f32(
                    false, a, false, b, (short)0, acc[t], false, false);
            }
        }
        __syncthreads();   // release this buffer before it is refilled
    }

    // ---- fused epilogue ----
    const int hi    = lane >> 4;                 // 0: rows v, 1: rows v+8
    const int rbase = n0 + wave * 16 + hi * 8;   // global row of acc[.][v]

    float rn[8], zz[8];
    #pragma unroll
    for (int v = 0; v < 8; ++v) { rn[v] = invn[rbase + v]; zz[v] = z2[rbase + v]; }
    float p2c[4];
    #pragma unroll
    for (int t = 0; t < 4; ++t) p2c[t] = p2[k0 + t * 16 + m];

    float rmv[8];
    #pragma unroll
    for (int v = 0; v < 8; ++v) rmv[v] = INFINITY;
    float cmv[4];
    #pragma unroll
    for (int t = 0; t < 4; ++t) cmv[t] = INFINITY;

    #pragma unroll
    for (int t = 0; t < 4; ++t) {
        const int gc = k0 + t * 16 + m;
        #pragma unroll
        for (int v = 0; v < 8; ++v) {
            const float dot = acc[t][v];
            xmapped[(size_t)(rbase + v) * K_PROT + gc] = dot * rn[v];
            const float sq = zz[v] + p2c[t] - 2.0f * dot;
            const float dist = sqrtf(fmaxf(sq, 0.0f));
            rmv[v] = fminf(rmv[v], dist);
            cmv[t] = fminf(cmv[t], dist);
        }
    }
    #pragma unroll
    for (int v = 0; v < 8; ++v) {
        float r = rmv[v];
        r = fminf(r, __shfl_xor(r, 1, 32));
        r = fminf(r, __shfl_xor(r, 2, 32));
        r = fminf(r, __shfl_xor(r, 4, 32));
        r = fminf(r, __shfl_xor(r, 8, 32));
        if (m == 0)
            atomicMin(&rowmin[rbase + v], __float_as_uint(r));
    }
    #pragma unroll
    for (int t = 0; t < 4; ++t) {
        float cmin = fminf(cmv[t], __shfl_xor(cmv[t], 16, 32));
        if (hi == 0)
            atomicMin(&colmin[k0 + t * 16 + m], __float_as_uint(cmin));
    }
}

// ---------------------------------------------------------------------------
// Kernel 4: means of rowmin/colmin, scalar losses
// ---------------------------------------------------------------------------
__global__ __launch_bounds__(256) void finalize_kernel(
    const unsigned int* __restrict__ rowmin, const unsigned int* __restrict__ colmin,
    const float* __restrict__ recon, const float* __restrict__ kl,
    const float* __restrict__ mmd,
    float* __restrict__ out_cvae, float* __restrict__ out_prot)
{
    __shared__ float sr[256], sc[256];
    const int tid = threadIdx.x;
    float a = 0.0f, b = 0.0f;
    for (int i = tid; i < N_ROWS; i += 256) a += __uint_as_float(rowmin[i]);
    for (int i = tid; i < K_PROT; i += 256) b += __uint_as_float(colmin[i]);
    sr[tid] = a; sc[tid] = b;
    __syncthreads();
    for (int s = 128; s > 0; s >>= 1) {
        if (tid < s) { sr[tid] += sr[tid + s]; sc[tid] += sc[tid + s]; }
        __syncthreads();
    }
    if (tid == 0) {
        const float p_min = sr[0] / (float)N_ROWS;
        const float f_min = sc[0] / (float)K_PROT;
        *out_prot = 0.5f * p_min + 0.5f * f_min;
        *out_cvae = recon[0] + 0.5f * kl[0] + mmd[0];
    }
}

// ---------------------------------------------------------------------------
extern "C" void kernel_launch(void* const* d_in, const int* in_sizes, int n_in,
                              void* d_out, int out_size, void* d_ws, size_t ws_size,
                              hipStream_t stream) {
    const float* Z     = (const float*)d_in[0];   // [N, D]
    const float* P     = (const float*)d_in[1];   // [K, D]
    const float* recon = (const float*)d_in[2];
    const float* kl    = (const float*)d_in[3];
    const float* mmd   = (const float*)d_in[4];

    float* out = (float*)d_out;
    float* out0 = out;                                         // encoder_out
    float* out1 = out + (size_t)N_ROWS * D_DIM;                // x
    float* out2 = out + 2 * (size_t)N_ROWS * D_DIM;            // x_mapped
    float* out_cvae = out2 + (size_t)N_ROWS * K_PROT;          // cvae_loss
    float* out_prot = out_cvae + 1;                            // prot_decoding_loss

    // workspace (floats): z2[N] | invn[N] | p2[K] | rowmin[N] | colmin[K]
    float* ws = (float*)d_ws;
    float* z2   = ws;
    float* invn = ws + N_ROWS;
    float* p2   = ws + 2 * N_ROWS;
    unsigned int* rowmin = (unsigned int*)(ws + 2 * N_ROWS + K_PROT);
    unsigned int* colmin = rowmin + N_ROWS;

    prep_kernel<<<K_PROT / 8, 256, 0, stream>>>(P, p2, colmin);
    rownorm_kernel<<<N_ROWS / 8, 256, 0, stream>>>(Z, out0, out1, z2, invn, rowmin);
    dim3 grid(N_ROWS / 128, K_PROT / 64);
    gemm_fused_kernel<<<grid, 256, 0, stream>>>(Z, P, z2, invn, p2, out2, rowmin, colmin);
    finalize_kernel<<<1, 256, 0, stream>>>(rowmin, colmin, recon, kl, mmd,
                                           out_cvae, out_prot);
}